// NGPNetwork_8289286881379
// MI455X (gfx1250) — compile-verified
//
#include <hip/hip_runtime.h>
#include <math.h>

// ---------------------------------------------------------------------------
// Instant-NGP fused forward for gfx1250 (MI455X).
//  - hashgrid encode (all 16 levels are DENSE-indexed: (R+1)^3 <= 2^19)
//  - sigma MLP 32->64->16, SH(16) encode, color MLP 31->64->64->1
//  - all GEMMs via V_WMMA_F32_16X16X4_F32 (fp32 tensor op, exact)
// 128 threads / block = 4 waves; 64 points / block (16-row M-tile per wave).
// ---------------------------------------------------------------------------

typedef float v2f __attribute__((ext_vector_type(2)));
typedef float v8f __attribute__((ext_vector_type(8)));

#define NLEV     16
#define TABLE_SZ (1 << 19)

struct ResArr { int r[NLEV]; };

// ---- WMMA helpers ---------------------------------------------------------
// fp32 WMMA 16x16x4 fragment layouts (ISA 7.12.2, wave32):
//   A (16x4): lane holds row M = lane&15; K = k0 + vgpr + 2*(lane>=16)
//   B (4x16): lane holds col N = lane&15; K = k0 + vgpr + 2*(lane>=16)
//   C/D     : lane holds col N = lane&15; row M = vgpr + 8*(lane>=16)

template<int KTILES>
__device__ __forceinline__ v8f gemm16_acc(const float* __restrict__ A, int lda,
                                          const float* __restrict__ B, int ldb,
                                          int lane)
{
    const int rc = lane & 15;            // A row / B col
    const int kk = (lane >> 4) << 1;     // 0 or 2
    v8f acc = {};
#pragma unroll
    for (int kt = 0; kt < KTILES; ++kt) {
        const int k = kt * 4 + kk;
        v2f a, b;
        a.x = A[rc * lda + k];
        a.y = A[rc * lda + k + 1];
        b.x = B[k * ldb + rc];
        b.y = B[(k + 1) * ldb + rc];
        acc = __builtin_amdgcn_wmma_f32_16x16x4_f32(
            /*neg_a=*/false, a, /*neg_b=*/false, b,
            /*c_mod=*/(short)0, acc, /*reuse_a=*/false, /*reuse_b=*/false);
    }
    return acc;
}

template<int KTILES, bool RELU>
__device__ __forceinline__ void gemm16(const float* __restrict__ A, int lda,
                                       const float* __restrict__ B, int ldb,
                                       float* __restrict__ D, int ldd, int lane)
{
    v8f acc = gemm16_acc<KTILES>(A, lda, B, ldb, lane);
    const int col = lane & 15;
    const int r0  = (lane >> 4) << 3;    // 0 or 8
#pragma unroll
    for (int i = 0; i < 8; ++i) {
        float v = acc[i];
        if (RELU) v = fmaxf(v, 0.0f);
        D[(r0 + i) * ldd + col] = v;
    }
}

// ---- fused kernel ---------------------------------------------------------

__global__ __launch_bounds__(128) void ngp_fused_kernel(
    const float* __restrict__ xin, const float* __restrict__ din,
    const float* __restrict__ emb,
    const float* __restrict__ ws0, const float* __restrict__ ws1,
    const float* __restrict__ wc0, const float* __restrict__ wc1,
    const float* __restrict__ wc2,
    float* __restrict__ out, ResArr res, int Ntot)
{
    // activations (strides padded to odd to dodge LDS bank conflicts)
    __shared__ float sA[64 * 33];   // hash feats (32 wide), later HD (sh|geo)
    __shared__ float sB[64 * 65];   // 64-wide intermediate (out1, then c1)
    __shared__ float sC[64 * 65];   // SH scratch in cols 0..15, later c2
    __shared__ float sH2[64 * 17];  // sigma-MLP output (16 wide)
    __shared__ float s_wc0[32 * 64];// wc0 zero-padded 31x64 -> 32x64
    __shared__ float s_wc2[64 * 16];// wc2 zero-padded 64x1  -> 64x16

    const int tid  = threadIdx.x;
    const int lane = tid & 31;
    const int wave = tid >> 5;

    // ---- stage padded weights --------------------------------------------
    for (int i = tid; i < 32 * 64; i += 128) {
        const int rr = i >> 6, cc = i & 63;
        s_wc0[i] = (rr < 31) ? wc0[rr * 64 + cc] : 0.0f;
    }
    for (int i = tid; i < 64 * 16; i += 128) {
        const int kkk = i >> 4, nn = i & 15;
        s_wc2[i] = (nn == 0) ? wc2[kkk] : 0.0f;
    }

    // ---- encode: 2 threads per point (8 levels each) ---------------------
    const int p    = tid >> 1;           // 0..63 local point
    const int ph   = tid & 1;            // which half of the levels
    const int gp   = blockIdx.x * 64 + p;
    const int gpc  = (gp < Ntot) ? gp : (Ntot - 1);

    const float px = (xin[gpc * 3 + 0] + 1.0f) * 0.5f;
    const float py = (xin[gpc * 3 + 1] + 1.0f) * 0.5f;
    const float pz = (xin[gpc * 3 + 2] + 1.0f) * 0.5f;

    if (ph == 0) {
        // SH degree-4 encode -> sC[p][0..15]
        const float dx0 = din[gpc * 3 + 0];
        const float dy0 = din[gpc * 3 + 1];
        const float dz0 = din[gpc * 3 + 2];
        const float xx = dx0 * dx0, yy = dy0 * dy0, zz = dz0 * dz0;
        const float xy = dx0 * dy0, yz = dy0 * dz0, xz = dx0 * dz0;
        float* sh = &sC[p * 65];
        sh[0]  = 0.28209479177387814f;
        sh[1]  = -0.48860251190291987f * dy0;
        sh[2]  =  0.48860251190291987f * dz0;
        sh[3]  = -0.48860251190291987f * dx0;
        sh[4]  =  1.0925484305920792f * xy;
        sh[5]  = -1.0925484305920792f * yz;
        sh[6]  =  0.94617469575756f * zz - 0.31539156525252005f;
        sh[7]  = -1.0925484305920792f * xz;
        sh[8]  =  0.5462742152960396f * (xx - yy);
        sh[9]  =  0.5900435899266435f * dy0 * (yy - 3.0f * xx);
        sh[10] =  2.890611442640554f * xy * dz0;
        sh[11] =  0.4570457994644657f * dy0 * (1.0f - 5.0f * zz);
        sh[12] =  0.3731763325901154f * dz0 * (5.0f * zz - 3.0f);
        sh[13] =  0.4570457994644657f * dx0 * (1.0f - 5.0f * zz);
        sh[14] =  1.445305721320277f * dz0 * (xx - yy);
        sh[15] =  0.5900435899266435f * dx0 * (3.0f * yy - xx);
    }

#pragma unroll
    for (int li = 0; li < 8; ++li) {
        const int l = ph * 8 + li;
        const int R = res.r[l];
        const float Rf = (float)R;
        const float gx = px * Rf, gy = py * Rf, gz = pz * Rf;
        int x0 = (int)floorf(gx); x0 = x0 < 0 ? 0 : (x0 > R - 1 ? R - 1 : x0);
        int y0 = (int)floorf(gy); y0 = y0 < 0 ? 0 : (y0 > R - 1 ? R - 1 : y0);
        int z0 = (int)floorf(gz); z0 = z0 < 0 ? 0 : (z0 > R - 1 ? R - 1 : z0);
        const float fx = gx - (float)x0;
        const float fy = gy - (float)y0;
        const float fz = gz - (float)z0;
        const int R1 = R + 1, R1sq = R1 * R1;
        const int base = x0 + y0 * R1 + z0 * R1sq;
        const float* eb = emb + (size_t)l * (size_t)(TABLE_SZ * 2);
        float f0 = 0.0f, f1 = 0.0f;
#pragma unroll
        for (int c = 0; c < 8; ++c) {
            const int cx = c & 1, cy = (c >> 1) & 1, cz = (c >> 2) & 1;
            const int idx = base + cx + cy * R1 + cz * R1sq;
            const float2 e = *(const float2*)(eb + 2 * idx);
            const float w = (cx ? fx : 1.0f - fx) *
                            (cy ? fy : 1.0f - fy) *
                            (cz ? fz : 1.0f - fz);
            f0 += e.x * w;
            f1 += e.y * w;
        }
        sA[p * 33 + 2 * l]     = f0;
        sA[p * 33 + 2 * l + 1] = f1;
    }
    __syncthreads();

    // ---- sigma MLP layer 1: relu(H[64x32] @ ws0[32x64]) -> sB ------------
    const int m0 = wave * 16;
#pragma unroll
    for (int nt = 0; nt < 4; ++nt)
        gemm16<8, true>(&sA[m0 * 33], 33, ws0 + nt * 16, 64,
                        &sB[m0 * 65 + nt * 16], 65, lane);
    __syncthreads();

    // ---- sigma MLP layer 2: out1[64x64] @ ws1[64x16] -> sH2 (no relu) ----
    gemm16<16, false>(&sB[m0 * 65], 65, ws1, 16, &sH2[m0 * 17], 17, lane);
    __syncthreads();

    // ---- sigma output + build HD = [sh(16) | geo(15) | 0] in sA ----------
    if (ph == 0) {
        if (gp < Ntot) out[gp] = fabsf(sH2[p * 17 + 0]);
#pragma unroll
        for (int j = 0; j < 16; ++j) sA[p * 33 + j] = sC[p * 65 + j];
    } else {
#pragma unroll
        for (int j = 0; j < 15; ++j) sA[p * 33 + 16 + j] = sH2[p * 17 + 1 + j];
        sA[p * 33 + 31] = 0.0f;
    }
    __syncthreads();

    // ---- color layer 1: relu(HD[64x32] @ wc0p[32x64]) -> sB --------------
#pragma unroll
    for (int nt = 0; nt < 4; ++nt)
        gemm16<8, true>(&sA[m0 * 33], 33, s_wc0 + nt * 16, 64,
                        &sB[m0 * 65 + nt * 16], 65, lane);
    __syncthreads();

    // ---- color layer 2: relu(c1[64x64] @ wc1[64x64]) -> sC ---------------
#pragma unroll
    for (int nt = 0; nt < 4; ++nt)
        gemm16<16, true>(&sB[m0 * 65], 65, wc1 + nt * 16, 64,
                        &sC[m0 * 65 + nt * 16], 65, lane);
    __syncthreads();

    // ---- color layer 3: c2[64x64] @ wc2p[64x16], col 0 = color -----------
    {
        v8f acc = gemm16_acc<16>(&sC[m0 * 65], 65, s_wc2, 16, lane);
        if ((lane & 15) == 0) {               // lanes 0 & 16 hold column N=0
            const int r0 = (lane >> 4) << 3;  // rows 0..7 / 8..15
#pragma unroll
            for (int i = 0; i < 8; ++i) {
                const int gi = blockIdx.x * 64 + m0 + r0 + i;
                if (gi < Ntot) out[Ntot + gi] = fabsf(acc[i]);
            }
        }
    }
}

// ---------------------------------------------------------------------------

extern "C" void kernel_launch(void* const* d_in, const int* in_sizes, int n_in,
                              void* d_out, int out_size, void* d_ws, size_t ws_size,
                              hipStream_t stream) {
    const float* x    = (const float*)d_in[0];
    const float* d    = (const float*)d_in[1];
    const float* emb  = (const float*)d_in[2];
    const float* ws0  = (const float*)d_in[3];
    const float* ws1  = (const float*)d_in[4];
    const float* wc0  = (const float*)d_in[5];
    const float* wc1  = (const float*)d_in[6];
    const float* wc2  = (const float*)d_in[7];
    float* out        = (float*)d_out;

    const int N = in_sizes[0] / 3;

    // RESOLUTIONS = floor(16 * s^l), s = exp(log(64/16)/15) — double, as numpy
    ResArr res;
    const double s = exp(log(64.0 / 16.0) / 15.0);
    for (int l = 0; l < NLEV; ++l)
        res.r[l] = (int)floor(16.0 * pow(s, (double)l));

    const int blocks = (N + 63) / 64;
    ngp_fused_kernel<<<blocks, 128, 0, stream>>>(
        x, d, emb, ws0, ws1, wc0, wc1, wc2, out, res, N);
}